// DistMultScore_76124000354698
// MI455X (gfx1250) — compile-verified
//
#include <hip/hip_runtime.h>
#include <stdint.h>

// DistMult edge scoring for MI455X (gfx1250).
// Memory-bound (AI ~0.6 FLOP/B): design is pure data movement.
//  - one wave32 per edge; lane l handles floats [4l, 4l+3] of the 128-float row
//  - rel_emb (307 MB, streamed once) -> async global->LDS copy with NT hint,
//    overlapped with the two random node_emb gathers (node_emb = 51 MB, kept
//    resident in the 192 MB L2 because the big stream bypasses it)
//  - wave32 shuffle-tree reduction, NT store of the score

typedef float float4v __attribute__((ext_vector_type(4)));

__global__ __launch_bounds__(256) void distmult_score_kernel(
    const float* __restrict__ node_emb,
    const float* __restrict__ rel_emb,
    const long long* __restrict__ src,
    const long long* __restrict__ dst,
    float* __restrict__ out,
    int n_edges)
{
    __shared__ float s_rel[8 * 128];        // 8 waves/block, 512 B per wave

    const int tid  = threadIdx.x;
    const int wave = tid >> 5;              // 0..7  (wave32)
    const int lane = tid & 31;              // 0..31
    const int e    = blockIdx.x * 8 + wave; // one edge per wave
    if (e >= n_edges) return;               // uniform per wave

    // ---- async-stage this wave's rel row into LDS (non-temporal stream) ----
    const float* rel_ptr = rel_emb + (size_t)e * 128u + (unsigned)(lane * 4);
    float*       lds_ptr = &s_rel[wave * 128 + lane * 4];
    unsigned     lds_off = (unsigned)(uintptr_t)lds_ptr;   // LDS byte offset (low 32b of flat shared addr)
    asm volatile("global_load_async_to_lds_b128 %0, %1, off th:TH_LOAD_NT"
                 :: "v"(lds_off), "v"(rel_ptr) : "memory");

    // ---- random gathers of head/tail rows (default policy: stay in L2) ----
    const long long s = __builtin_nontemporal_load(&src[e]);
    const long long d = __builtin_nontemporal_load(&dst[e]);
    const float4v h = *(const float4v*)(node_emb + (size_t)s * 128u + (unsigned)(lane * 4));
    const float4v t = *(const float4v*)(node_emb + (size_t)d * 128u + (unsigned)(lane * 4));

    // ---- consume the async-staged rel row ----
    asm volatile("s_wait_asynccnt 0x0" ::: "memory");
    const float4v r = *(const float4v*)lds_ptr;

    // per-lane partial over 4 elements
    float p =       h.x * r.x * t.x;
    p = fmaf(h.y * r.y, t.y, p);
    p = fmaf(h.z * r.z, t.z, p);
    p = fmaf(h.w * r.w, t.w, p);

    // wave32 tree reduction (5 steps)
    p += __shfl_xor(p, 16, 32);
    p += __shfl_xor(p,  8, 32);
    p += __shfl_xor(p,  4, 32);
    p += __shfl_xor(p,  2, 32);
    p += __shfl_xor(p,  1, 32);

    if (lane == 0)
        __builtin_nontemporal_store(p, &out[e]);
}

extern "C" void kernel_launch(void* const* d_in, const int* in_sizes, int n_in,
                              void* d_out, int out_size, void* d_ws, size_t ws_size,
                              hipStream_t stream) {
    const float*     node_emb = (const float*)d_in[0];
    const float*     rel_emb  = (const float*)d_in[1];
    const long long* src      = (const long long*)d_in[2];   // jnp.int64
    const long long* dst      = (const long long*)d_in[3];   // jnp.int64
    float*           out      = (float*)d_out;

    const int n_edges = in_sizes[2];
    const int blocks  = (n_edges + 7) / 8;   // 8 edges (waves) per 256-thread block
    distmult_score_kernel<<<blocks, 256, 0, stream>>>(node_emb, rel_emb, src, dst, out, n_edges);
}